// STFT_12060268167739
// MI455X (gfx1250) — compile-verified
//
#include <hip/hip_runtime.h>

// ---------------------------------------------------------------------------
// STFT as WMMA-f32 GEMM for gfx1250 (MI455X).
//   signal: (16, 480000) f32      basis: (1026, 1, 1024) f32
//   out   : magnitude (16,513,1876) ++ phase (16,513,1876), f32
// Per wave: one 16-row freq tile x two 16-frame tiles, real+imag accumulators
// (4x v8f). Inner loop: 2 global_b64 (basis) + 2 ds_b64 (frames) + 4 WMMA.
// Interior blocks fill LDS with global_load_async_to_lds_b128 (ASYNCcnt).
// ---------------------------------------------------------------------------

#define SIG_LEN     480000
#define BATCH       16
#define WIN         1024
#define HOP         256
#define PAD         512
#define CUTOFF      513            // FFT_SIZE/2 + 1
#define NFRAMES     1876           // 480000/256 + 1
#define FPB         32             // frames per block (2 x 16-frame tiles)
#define LDS_STRIDE  1028           // 4112 B: 16B-aligned rows, conflict-free banks
#define M_TILES     33             // ceil(513/16)

typedef __attribute__((ext_vector_type(2))) float v2f;
typedef __attribute__((ext_vector_type(8))) float v8f;
typedef __attribute__((ext_vector_type(4))) int   v4i;

#if defined(__gfx1250__) && __has_builtin(__builtin_amdgcn_global_load_async_to_lds_b128)
#define HAVE_ASYNC_LDS 1
typedef __attribute__((address_space(1))) v4i* gas_v4i;   // global (AS1) int4*
typedef __attribute__((address_space(3))) v4i* lds_v4i;   // LDS    (AS3) int4*
#else
#define HAVE_ASYNC_LDS 0
#endif

__device__ __forceinline__ int reflect_idx(int s) {
    // np.pad(..., mode='reflect'): -1 -> 1 ; SIG_LEN -> SIG_LEN-2
    s = (s < 0) ? -s : s;
    s = (s >= SIG_LEN) ? (2 * SIG_LEN - 2 - s) : s;
    return s;
}

__global__ __launch_bounds__(256)
void stft_wmma_f32(const float* __restrict__ signal,
                   const float* __restrict__ basis,
                   float* __restrict__ out)
{
    extern __shared__ float sF[];            // FPB frames x LDS_STRIDE floats

    const int b       = blockIdx.z;
    const int t_blk   = blockIdx.x;
    const int m_group = blockIdx.y;
    const int tid     = threadIdx.x;
    const int t0      = t_blk * FPB;

    const float* __restrict__ sig_b = signal + (size_t)b * SIG_LEN;

    // ---- Stage 32 overlapping frames into LDS ------------------------------
    const bool interior = (t0 * HOP - PAD >= 0) &&
                          ((t0 + FPB - 1) * HOP + (WIN - 1) - PAD < SIG_LEN);
    if (interior) {
        const int base = t0 * HOP - PAD;     // first padded sample, in-bounds
        // FPB*WIN/4 = 8192 float4 copies, 32 per thread
        for (int j = tid; j < FPB * (WIN / 4); j += 256) {
            int fr  = j >> 8;                // frame within block
            int kq  = j & 255;               // float4 index within window
            int go  = base + fr * HOP + kq * 4;
            int lo  = fr * LDS_STRIDE + kq * 4;
#if HAVE_ASYNC_LDS
            __builtin_amdgcn_global_load_async_to_lds_b128(
                (gas_v4i)(sig_b + go), (lds_v4i)(sF + lo), 0, 0);
#else
            *(float4*)(sF + lo) = *(const float4*)(sig_b + go);
#endif
        }
#if HAVE_ASYNC_LDS
#if __has_builtin(__builtin_amdgcn_s_wait_asynccnt)
        __builtin_amdgcn_s_wait_asynccnt(0);
#else
        asm volatile("s_wait_asynccnt 0x0" ::: "memory");
#endif
#endif
    } else {
        // boundary blocks: scalar copy with reflect mapping
        #pragma unroll 4
        for (int idx = tid; idx < FPB * WIN; idx += 256) {
            int fr = idx >> 10;
            int k  = idx & (WIN - 1);
            int s  = reflect_idx((t0 + fr) * HOP + k - PAD);
            sF[fr * LDS_STRIDE + k] = sig_b[s];
        }
    }
    __syncthreads();

    // ---- Per-wave: 16 freq rows x 32 frames, real+imag ---------------------
    const int wave   = tid >> 5;
    const int lane   = tid & 31;
    const int m_tile = m_group * 8 + wave;
    if (m_tile >= M_TILES) return;           // wave-uniform; no barrier after
    const int m0 = m_tile * 16;

    int mrow = m0 + (lane & 15);
    if (mrow > CUTOFF - 1) mrow = CUTOFF - 1;        // clamp pad rows (stores masked)
    const int koff = (lane >> 4) * 2;                // lanes 0-15: K{0,1}; 16-31: K{2,3}

    const float* __restrict__ aRe = basis + (size_t)mrow * WIN + koff;
    const float* __restrict__ aIm = basis + (size_t)(mrow + CUTOFF) * WIN + koff;
    const float* __restrict__ bL0 = sF + (lane & 15) * LDS_STRIDE + koff;
    const float* __restrict__ bL1 = bL0 + 16 * LDS_STRIDE;

    v8f accRe0 = {}, accIm0 = {};
    v8f accRe1 = {}, accIm1 = {};

    #pragma unroll 4
    for (int kk = 0; kk < WIN; kk += 4) {
        v2f a_re = *(const v2f*)(aRe + kk);          // global_load_b64 (L2-hot)
        v2f a_im = *(const v2f*)(aIm + kk);
        v2f b0   = *(const v2f*)(bL0 + kk);          // ds_load_b64, conflict-free
        v2f b1   = *(const v2f*)(bL1 + kk);
        accRe0 = __builtin_amdgcn_wmma_f32_16x16x4_f32(
                     false, a_re, false, b0, (short)0, accRe0, false, false);
        accIm0 = __builtin_amdgcn_wmma_f32_16x16x4_f32(
                     false, a_im, false, b0, (short)0, accIm0, false, false);
        accRe1 = __builtin_amdgcn_wmma_f32_16x16x4_f32(
                     false, a_re, false, b1, (short)0, accRe1, false, false);
        accIm1 = __builtin_amdgcn_wmma_f32_16x16x4_f32(
                     false, a_im, false, b1, (short)0, accIm1, false, false);
    }

    // ---- Epilogue: magnitude + phase, coalesced along t --------------------
    const size_t MT  = (size_t)CUTOFF * NFRAMES;
    float* __restrict__ magP = out + (size_t)b * MT;
    float* __restrict__ phP  = out + (size_t)BATCH * MT + (size_t)b * MT;

    const int t_a   = t0 + (lane & 15);
    const int t_b   = t_a + 16;
    const int mbase = m0 + ((lane >> 4) << 3);       // C/D: M = r + 8*(lane>=16)

    #pragma unroll
    for (int r = 0; r < 8; ++r) {
        int m = mbase + r;
        if (m < CUTOFF) {
            size_t row = (size_t)m * NFRAMES;
            if (t_a < NFRAMES) {
                float re = accRe0[r], im = accIm0[r];
                magP[row + t_a] = sqrtf(re * re + im * im);
                phP[row + t_a]  = atan2f(im, re);
            }
            if (t_b < NFRAMES) {
                float re = accRe1[r], im = accIm1[r];
                magP[row + t_b] = sqrtf(re * re + im * im);
                phP[row + t_b]  = atan2f(im, re);
            }
        }
    }
}

extern "C" void kernel_launch(void* const* d_in, const int* in_sizes, int n_in,
                              void* d_out, int out_size, void* d_ws, size_t ws_size,
                              hipStream_t stream)
{
    const float* signal = (const float*)d_in[0];   // (16, 480000) f32
    const float* basis  = (const float*)d_in[1];   // (1026, 1, 1024) f32
    float* out          = (float*)d_out;           // mag ++ phase, f32

    dim3 grid((NFRAMES + FPB - 1) / FPB,           // 59 frame blocks
              (M_TILES + 7) / 8,                   // 5 groups of 8 m-tiles
              BATCH);                              // 16
    dim3 block(256);                               // 8 waves (wave32)
    size_t shmem = (size_t)FPB * LDS_STRIDE * sizeof(float);  // ~128.5 KB

    stft_wmma_f32<<<grid, block, shmem, stream>>>(signal, basis, out);
}